// MHAWrapper_19404662244044
// MI455X (gfx1250) — compile-verified
//
#include <hip/hip_runtime.h>
#include <hip/hip_bf16.h>

// MI455X (gfx1250) decode-step MHA.
//
// Roofline: KV-cache shift dominates (2 x 335 MB read + 2 x 335 MB write
// ~= 1.34 GB -> ~59 us at 23.3 TB/s). Total FLOPs ~0.75 GF -> memory bound.
// Strategy: fuse the cache shift with the attention GEMVs (touch K/V once),
// and run the four 32x1280x1280 projections on V_WMMA_F32_16X16X4_F32
// (native fp32 WMMA -> exact precision match with the fp32 reference).
//
// V slice ([ctx, dh], shift along ctx) keeps rows contiguous -> b128 per lane.
// K slice ([dh, ctx], +1 shift along the contiguous axis) -> coalesced b32
// (vectorizing would either misalign loads or de-coalesce them).

typedef __attribute__((ext_vector_type(2))) float v2f;
typedef __attribute__((ext_vector_type(4))) float v4f;
typedef __attribute__((ext_vector_type(8))) float v8f;

#define D_MODEL 1280
#define N_BATCH 32
#define N_HEAD  20
#define DHEAD   64
#define CTX     2048

// ---------------------------------------------------------------------------
// C[32,1280] = A[32,K] @ W[1280,K]^T (+ bias)   via V_WMMA_F32_16X16X4_F32
//
// Block = 64 threads = 2 waves. Wave w computes rows [16w, 16w+16) of C for
// the 16-column tile n0 = blockIdx.x*16.
//
// f32 WMMA operand layouts (ISA 7.12.2):
//   A 16x4 : VGPR0 = K=0 (lanes 0-15) / K=2 (lanes 16-31); VGPR1 = K=1 / K=3
//            -> lane holds float2 {A[m, k0+koff], A[m, k0+koff+1]}, koff=2*(lane>>4)
//   B 4x16 : symmetric (rows striped across lanes within a VGPR)
//            -> lane holds float2 {Wt[k0+koff, n], Wt[k0+koff+1, n]} = W[n, k0+koff..]
//   C 16x16: VGPR r = row r (lanes 0-15) / row 8+r (lanes 16-31)
// ---------------------------------------------------------------------------
__global__ __launch_bounds__(64) void gemm32_wmma_f32(
    const float* __restrict__ A,     // [32, K] row-major
    const float* __restrict__ W,     // [1280, K] row-major; computes A @ W^T
    const float* __restrict__ bias,  // [1280] or nullptr
    float* __restrict__ C,           // [32, 1280] row-major
    int K)
{
  const int lane   = threadIdx.x & 31;
  const int m_base = (threadIdx.x >> 5) * 16;   // wave 0 -> rows 0..15, wave 1 -> 16..31
  const int n0     = blockIdx.x * 16;
  const int half   = lane >> 4;                 // 0: lanes 0-15, 1: lanes 16-31
  const int l15    = lane & 15;
  const int koff   = half * 2;

  const float* arow = A + (size_t)(m_base + l15) * K + koff;   // A row m, K stripe
  const float* wrow = W + (size_t)(n0 + l15) * K + koff;       // W row n, K stripe

  v8f c = {0.f, 0.f, 0.f, 0.f, 0.f, 0.f, 0.f, 0.f};
#pragma unroll 4
  for (int k0 = 0; k0 < K; k0 += 4) {
    v2f a = *(const v2f*)(arow + k0);   // 8B aligned: k0 % 4 == 0, koff even
    v2f b = *(const v2f*)(wrow + k0);
    c = __builtin_amdgcn_wmma_f32_16x16x4_f32(
        /*neg_a=*/false, a, /*neg_b=*/false, b,
        /*c_mod=*/(short)0, c, /*reuse_a=*/false, /*reuse_b=*/false);
  }

  const float bn = bias ? bias[n0 + l15] : 0.0f;
#pragma unroll
  for (int r = 0; r < 8; ++r) {
    const int m = m_base + half * 8 + r;          // C/D layout: M=r / M=8+r
    C[(size_t)m * D_MODEL + n0 + l15] = c[r] + bn;
  }
}

// ---------------------------------------------------------------------------
// Fused per-(b,h) kernel: rolling cache shift + attention, single pass over
// K and V slices (read-once / write-once -> bandwidth roofline).
//   k_out[d, j]  = k_cache[d, j+1] (j<2047) | k_new[d]  ; logits[j] = q . k[:,j]
//   softmax over 2048 logits (LDS)
//   v_out[j, :]  = v_cache[j+1, :] (j<2047) | v_new[:]  ; wv = sum_j w[j] v[j,:]
// 640 blocks (b*h), 256 threads (8 waves).
// ---------------------------------------------------------------------------
__global__ __launch_bounds__(256) void attn_shift_kernel(
    const float* __restrict__ q_ws,    // [32*1280] = [bh*64 + d]
    const float* __restrict__ kn_ws,   // [32*1280]
    const float* __restrict__ vn_ws,   // [32*1280]
    const float* __restrict__ k_cache, // [b,h,dh,ctx]
    const float* __restrict__ v_cache, // [b,h,ctx,dh]
    const float* __restrict__ mask,    // [ctx]
    float* __restrict__ k_out,         // [b,h,dh,ctx]
    float* __restrict__ v_out,         // [b,h,ctx,dh]
    float* __restrict__ wv_ws)         // [32*1280]
{
  const int bh = blockIdx.x;           // b*20 + h; note b*1280 + h*64 == bh*64
  const int t  = threadIdx.x;

  __shared__ float sQ[DHEAD];
  __shared__ float sL[CTX];            // logits -> softmax weights (8 KB)
  __shared__ float sRed[256];
  __shared__ float sPart[16 * DHEAD];  // V-pass partials (4 KB)

  const float* kc = k_cache + (size_t)bh * DHEAD * CTX;
  const float* vc = v_cache + (size_t)bh * CTX * DHEAD;
  float*       ko = k_out   + (size_t)bh * DHEAD * CTX;
  float*       vo = v_out   + (size_t)bh * CTX * DHEAD;

  if (t < DHEAD) sQ[t] = q_ws[(size_t)bh * DHEAD + t];
  __syncthreads();

  // ---- K pass: shift-copy + q.k logits. Thread t owns ctx cols {t + 256*i}.
  // Coalesced b32: 32 lanes x 4B = one 128B line per instruction; the +1 shift
  // on the contiguous axis forbids aligned wide accesses here.
  float acc[8] = {0.f, 0.f, 0.f, 0.f, 0.f, 0.f, 0.f, 0.f};
  for (int d = 0; d < DHEAD; ++d) {
    const float qd = sQ[d];
    const float* krow  = kc + (size_t)d * CTX;
    float*       korow = ko + (size_t)d * CTX;
    if (d + 1 < DHEAD) __builtin_prefetch(kc + (size_t)(d + 1) * CTX + t, 0, 0);
#pragma unroll
    for (int i = 0; i < 8; ++i) {
      const int j = t + i * 256;
      const float kv = (j < CTX - 1) ? krow[j + 1]
                                     : kn_ws[(size_t)bh * DHEAD + d];
      korow[j] = kv;
      acc[i] += qd * kv;
    }
  }
#pragma unroll
  for (int i = 0; i < 8; ++i) {
    const int j = t + i * 256;
    sL[j] = acc[i] + mask[j];
  }
  __syncthreads();

  // ---- softmax over CTX logits ----
  float m = -INFINITY;
#pragma unroll
  for (int i = 0; i < 8; ++i) m = fmaxf(m, sL[t + i * 256]);
  sRed[t] = m;
  __syncthreads();
  for (int s = 128; s > 0; s >>= 1) {
    if (t < s) sRed[t] = fmaxf(sRed[t], sRed[t + s]);
    __syncthreads();
  }
  const float gmax = sRed[0];
  __syncthreads();

  float lsum = 0.f;
#pragma unroll
  for (int i = 0; i < 8; ++i) {
    const int j = t + i * 256;
    const float e = __expf(sL[j] - gmax);   // v_exp_f32
    sL[j] = e;
    lsum += e;
  }
  __syncthreads();
  sRed[t] = lsum;
  __syncthreads();
  for (int s = 128; s > 0; s >>= 1) {
    if (t < s) sRed[t] += sRed[t + s];
    __syncthreads();
  }
  const float inv = 1.0f / sRed[0];
  __syncthreads();
#pragma unroll
  for (int i = 0; i < 8; ++i) sL[t + i * 256] *= inv;
  __syncthreads();

  // ---- V pass: shift-copy + w.V, fully b128 (rows contiguous, 16B aligned).
  // Thread t = (rowgroup rg = t>>4 of 16 ctx rows in flight, 4-col group c4).
  // Each half-wave streams one full 256B row per iteration.
  {
    const int c4 = (t & 15) * 4;
    const int rg = t >> 4;                       // 0..15
    const v4f vnv = *(const v4f*)(vn_ws + (size_t)bh * DHEAD + c4);
    v4f wacc = {0.f, 0.f, 0.f, 0.f};
    for (int j = rg; j < CTX; j += 16) {
      v4f vv;
      if (j < CTX - 1) {
        vv = *(const v4f*)(vc + (size_t)(j + 1) * DHEAD + c4);
      } else {
        vv = vnv;                                // only rg==15, final iteration
      }
      *(v4f*)(vo + (size_t)j * DHEAD + c4) = vv;
      wacc += sL[j] * vv;
      if (j + 17 < CTX)
        __builtin_prefetch(vc + (size_t)(j + 17) * DHEAD + c4, 0, 0);
    }
    __syncthreads();
    *(v4f*)(sPart + rg * DHEAD + c4) = wacc;
    __syncthreads();
    if (t < DHEAD) {
      float s = 0.f;
#pragma unroll
      for (int r = 0; r < 16; ++r) s += sPart[r * DHEAD + t];
      wv_ws[(size_t)bh * DHEAD + t] = s;
    }
  }
}

// ---------------------------------------------------------------------------
extern "C" void kernel_launch(void* const* d_in, const int* in_sizes, int n_in,
                              void* d_out, int out_size, void* d_ws, size_t ws_size,
                              hipStream_t stream) {
  const float* x       = (const float*)d_in[0];   // [32,1,1280]
  const float* mask    = (const float*)d_in[1];   // [1,1,1,2048]
  const float* k_cache = (const float*)d_in[2];   // [32,20,64,2048]
  const float* v_cache = (const float*)d_in[3];   // [32,20,2048,64]
  const float* Wq      = (const float*)d_in[4];
  const float* bq      = (const float*)d_in[5];
  const float* Wk      = (const float*)d_in[6];
  const float* Wv      = (const float*)d_in[7];
  const float* bv      = (const float*)d_in[8];
  const float* Wo      = (const float*)d_in[9];
  const float* bo      = (const float*)d_in[10];

  float* out   = (float*)d_out;                         // [32,1280]
  float* k_out = out + (size_t)N_BATCH * D_MODEL;       // [32,20,64,2048]
  float* v_out = k_out + (size_t)N_BATCH * N_HEAD * DHEAD * CTX;

  float* ws    = (float*)d_ws;                          // 640 KB total
  float* q_ws  = ws;                                    // [32,1280]
  float* kn_ws = ws + (size_t)N_BATCH * D_MODEL;        // [32,1280]
  float* vn_ws = ws + (size_t)2 * N_BATCH * D_MODEL;    // [32,1280]
  float* wv_ws = ws + (size_t)3 * N_BATCH * D_MODEL;    // [32,1280]

  const dim3 gblk(D_MODEL / 16);  // 80 tiles of N
  const dim3 gthr(64);            // 2 waves

  // Q/K/V projections (independent; stream-ordered before attention).
  gemm32_wmma_f32<<<gblk, gthr, 0, stream>>>(x, Wq, bq,      q_ws,  D_MODEL);
  gemm32_wmma_f32<<<gblk, gthr, 0, stream>>>(x, Wk, nullptr, kn_ws, D_MODEL);
  gemm32_wmma_f32<<<gblk, gthr, 0, stream>>>(x, Wv, bv,      vn_ws, D_MODEL);

  // Fused cache-shift + attention: the 1.34 GB bandwidth-bound phase.
  attn_shift_kernel<<<N_BATCH * N_HEAD, 256, 0, stream>>>(
      q_ws, kn_ws, vn_ws, k_cache, v_cache, mask, k_out, v_out, wv_ws);

  // Output projection.
  gemm32_wmma_f32<<<gblk, gthr, 0, stream>>>(wv_ws, Wo, bo, out, D_MODEL);
}